// LocalContextNorm_18957985644696
// MI455X (gfx1250) — compile-verified
//
#include <hip/hip_runtime.h>

// ---------------------------------------------------------------------------
// LocalContextNorm, fused single kernel for MI455X (gfx1250).
//
// Memory-bound problem: ~536 MB min traffic -> ~23 us at 23.3 TB/s.
// Design: one WG per (n, group, 128-output-row chunk). Each of 512 threads
// owns one column; vertical 227-row window sums (VS,VQ) live in registers and
// are updated incrementally. Horizontal 227-window sums come from a 512-wide
// LDS/shuffle prefix scan (S[i][j] = P[j+227]-P[j]). Stat rows are edge-
// clamped to 512 wide and consumed immediately to normalize output rows.
// Rows are staged into LDS with gfx1250 async global->LDS DMA (ASYNCcnt),
// software-pipelined so the DMA for stat row i+1 overlaps the scan/normalize
// work of stat row i.
// ---------------------------------------------------------------------------

#define LCN_W     512
#define LCN_H     512
#define LCN_HW    (LCN_W * LCN_H)
#define LCN_C     32
#define LCN_G     16
#define LCN_WIN   227
#define LCN_SH    (LCN_H - LCN_WIN)   // 285 stat rows/cols
#define LCN_PT    113                 // left/top replicate pad
#define LCN_NCHUNK 4
#define LCN_RPC   (LCN_H / LCN_NCHUNK) // 128 output rows per chunk
#define LCN_INV_N (1.0f / (227.0f * 227.0f * 2.0f))
#define LCN_EPS   1e-5f

#ifndef __has_builtin
#define __has_builtin(x) 0
#endif

// clang prototype (probe-confirmed in round 1): async-LDS builtins take
// pointers to 16-byte int vectors, global (AS1) source and LDS (AS3) dest.
typedef int v4i __attribute__((ext_vector_type(4)));
typedef __attribute__((address_space(1))) v4i GV4;   // 16B block in global
typedef __attribute__((address_space(3))) v4i LV4;   // 16B block in LDS

#if __has_builtin(__builtin_amdgcn_global_load_async_to_lds_b128) && \
    __has_builtin(__builtin_amdgcn_s_wait_asynccnt)
// gfx1250 async global->LDS DMA path (tracked with ASYNCcnt)
#define ASYNC_COPY16(ldsp, gsrc) \
  __builtin_amdgcn_global_load_async_to_lds_b128((GV4*)(gsrc), (LV4*)(ldsp), 0, 0)
#define ASYNC_WAIT(n) __builtin_amdgcn_s_wait_asynccnt(n)
#else
// portable fallback: synchronous 16B copy through VGPRs
#define ASYNC_COPY16(ldsp, gsrc) (*(float4*)(ldsp) = *(const float4*)(gsrc))
#define ASYNC_WAIT(n) ((void)0)
#endif

// Stage one (row, both channels) pair: 1024 floats -> buf[0..511]=ch0, [512..]=ch1.
// Issued by threads 0..255, 16B per lane.
__device__ __forceinline__ void issue_row(float* buf, const float* base, int r) {
  const int t = threadIdx.x;
  if (t < 256) {
    const int ch  = t >> 7;            // 0..1
    const int col = (t & 127) << 2;    // 0,4,...,508
    ASYNC_COPY16(buf + ch * LCN_W + col,
                 base + (size_t)ch * LCN_HW + (size_t)r * LCN_W + col);
  }
}

// Stage lead row (i+227) into bufLead and trail row (i) into bufTrail,
// all 512 threads issue concurrently.
__device__ __forceinline__ void issue_two(float* bufLead, float* bufTrail,
                                          const float* base, int rl, int rt) {
  const int t = threadIdx.x;
  float* buf  = (t < 256) ? bufLead : bufTrail;
  const int r = (t < 256) ? rl : rt;
  const int tt  = t & 255;
  const int ch  = tt >> 7;
  const int col = (tt & 127) << 2;
  ASYNC_COPY16(buf + ch * LCN_W + col,
               base + (size_t)ch * LCN_HW + (size_t)r * LCN_W + col);
}

// Given this thread's vertical window sums (VS,VQ) for its column, build the
// 512-wide mean / inv-std row in LDS for the current stat row.
__device__ __forceinline__ void compute_stat_row(float VS, float VQ,
                                                 float* pref, float* wS, float* wQ,
                                                 float* mrow, float* irow) {
  const int t = threadIdx.x;
  const int lane = t & 31;
  const int wid  = t >> 5;

  // wave32 inclusive scan of VS/VQ across columns
  float ls = VS, lq = VQ;
#pragma unroll
  for (int d = 1; d < 32; d <<= 1) {
    float us = __shfl_up(ls, d, 32);
    float uq = __shfl_up(lq, d, 32);
    if (lane >= d) { ls += us; lq += uq; }
  }
  if (lane == 31) { wS[wid] = ls; wQ[wid] = lq; }
  __syncthreads();
  if (wid == 0) {
    float vs = (lane < 16) ? wS[lane] : 0.0f;
    float vq = (lane < 16) ? wQ[lane] : 0.0f;
#pragma unroll
    for (int d = 1; d < 16; d <<= 1) {
      float us = __shfl_up(vs, d, 32);
      float uq = __shfl_up(vq, d, 32);
      if (lane >= d) { vs += us; vq += uq; }
    }
    if (lane < 16) { wS[lane] = vs; wQ[lane] = vq; }
  }
  __syncthreads();
  const float offS = wid ? wS[wid - 1] : 0.0f;
  const float offQ = wid ? wQ[wid - 1] : 0.0f;
  pref[t]         = ls + offS;   // inclusive prefix P[c], cols 0..511
  pref[LCN_W + t] = lq + offQ;
  __syncthreads();

  // expand stat col j = clamp(ww-113, 0, 284); window = cols j+1..j+227
  int j = t - LCN_PT;
  j = j < 0 ? 0 : (j > LCN_SH - 1 ? LCN_SH - 1 : j);
  const float S = pref[j + LCN_WIN] - pref[j];
  const float Q = pref[LCN_W + j + LCN_WIN] - pref[LCN_W + j];
  const float mean = S * LCN_INV_N;
  const float var  = (Q - S * S * LCN_INV_N) * LCN_INV_N;
  mrow[t] = mean;
  irow[t] = rsqrtf(var + LCN_EPS);
  __syncthreads();
}

__global__ __launch_bounds__(512) void lcn_fused(const float* __restrict__ x,
                                                 const float* __restrict__ wgt,
                                                 const float* __restrict__ bia,
                                                 float* __restrict__ out) {
  __shared__ float bufA[2 * LCN_W];   // staging: [0..511]=ch0 row, [512..]=ch1 row
  __shared__ float bufB[2 * LCN_W];
  __shared__ float pref[2 * LCN_W];   // prefix sums for S and Q
  __shared__ float wS[16], wQ[16];
  __shared__ float mrow[LCN_W], irow[LCN_W];

  const int t = threadIdx.x;

  const int blk   = blockIdx.x;
  const int chunk = blk & (LCN_NCHUNK - 1);
  const int g     = (blk >> 2) & (LCN_G - 1);
  const int n     = blk >> 6;

  const float* base = x   + ((size_t)n * LCN_C + (size_t)g * 2) * LCN_HW;
  float* obase      = out + ((size_t)n * LCN_C + (size_t)g * 2) * LCN_HW;
  const float w0 = wgt[g * 2 + 0], w1 = wgt[g * 2 + 1];
  const float b0 = bia[g * 2 + 0], b1 = bia[g * 2 + 1];

  const int h0 = chunk * LCN_RPC;
  const int h1 = h0 + LCN_RPC;
  int icur = h0 - LCN_PT;
  icur = icur < 0 ? 0 : (icur > LCN_SH - 1 ? LCN_SH - 1 : icur);
  int imax = h1 - 1 - LCN_PT;                       // last stat row this chunk reaches
  imax = imax < 0 ? 0 : (imax > LCN_SH - 1 ? LCN_SH - 1 : imax);

  // ---- init: vertical window sums over rows icur+1 .. icur+227 ----
  // (row 0 is never part of any window -- integral-image convention)
  float VS = 0.0f, VQ = 0.0f;
  issue_row(bufA, base, icur + 1);
  for (int k = 0; k < LCN_WIN; ++k) {
    float* cur = (k & 1) ? bufB : bufA;
    if (k + 1 < LCN_WIN) {
      float* nxt = (k & 1) ? bufA : bufB;
      issue_row(nxt, base, icur + 2 + k);   // prefetch next row-pair
      ASYNC_WAIT(1);                        // cur's DMA done (in-order)
    } else {
      ASYNC_WAIT(0);
    }
    __syncthreads();
    const float a = cur[t];
    const float c = cur[LCN_W + t];
    VS += a + c;
    VQ += a * a + c * c;
    __syncthreads();                        // all done reading before re-issue
  }

  // Prefetch the first stat advance's lead/trail rows; they fly while we build
  // the first stat row and normalize the leading (edge-replicated) output rows.
  if (icur < imax)
    issue_two(bufA, bufB, base, icur + 1 + LCN_WIN, icur + 1);

  compute_stat_row(VS, VQ, pref, wS, wQ, mrow, irow);

  // ---- stream over this chunk's output rows ----
  for (int hh = h0; hh < h1; ++hh) {
    int ineed = hh - LCN_PT;
    ineed = ineed < 0 ? 0 : (ineed > LCN_SH - 1 ? LCN_SH - 1 : ineed);
    if (ineed != icur) {
      icur = ineed;                         // always advances by exactly 1
      ASYNC_WAIT(0);                        // prefetched rows have landed
      __syncthreads();
      const float la = bufA[t], lc = bufA[LCN_W + t];   // lead row icur+227
      const float ta = bufB[t], tc = bufB[LCN_W + t];   // trail row icur
      VS += (la + lc) - (ta + tc);
      VQ += (la * la + lc * lc) - (ta * ta + tc * tc);
      __syncthreads();                      // all reads done before re-issue
      if (icur < imax)                      // prefetch next advance's rows;
        issue_two(bufA, bufB, base,         // DMA overlaps scan + normalize
                  icur + 1 + LCN_WIN, icur + 1);
      compute_stat_row(VS, VQ, pref, wS, wQ, mrow, irow);
    }
    // normalize output row hh for both channels of the group
    const size_t o = (size_t)hh * LCN_W + t;
    const float a = base[o];
    const float c = base[LCN_HW + o];
    const float m = mrow[t];
    const float iv = irow[t];
    const float r0 = (a - m) * iv * w0 + b0;
    const float r1 = (c - m) * iv * w1 + b1;
    __builtin_nontemporal_store(r0, obase + o);           // streaming stores
    __builtin_nontemporal_store(r1, obase + LCN_HW + o);
  }
}

extern "C" void kernel_launch(void* const* d_in, const int* in_sizes, int n_in,
                              void* d_out, int out_size, void* d_ws, size_t ws_size,
                              hipStream_t stream) {
  (void)in_sizes; (void)n_in; (void)d_ws; (void)ws_size; (void)out_size;
  const float* x = (const float*)d_in[0];
  const float* w = (const float*)d_in[1];
  const float* b = (const float*)d_in[2];
  float* out = (float*)d_out;
  dim3 grid(8 * LCN_G * LCN_NCHUNK);   // N * groups * row-chunks = 512 WGs
  dim3 block(512);                     // 16 wave32s, one column per thread
  hipLaunchKernelGGL(lcn_fused, grid, block, 0, stream, x, w, b, out);
}